// AttentiveTransformer_90795608637740
// MI455X (gfx1250) — compile-verified
//
#include <hip/hip_runtime.h>
#include <hip/hip_bf16.h>

typedef __attribute__((ext_vector_type(2))) float v2f;
typedef __attribute__((ext_vector_type(8))) float v8f;

#define B_TOTAL 65536
#define K_IN    128
#define N_OUT   512
#define VBATCH  128

// ---------------------------------------------------------------------------
// Kernel 1: h = x @ W^T + bias   (h written into d_out, used as scratch; it
// is 128 MB so it stays resident in the 192 MB L2 for kernel 2).
//
// Block = one 128-row virtual batch (512 blocks), 256 threads = 8 waves.
// Wave w computes rows [w*16, w*16+16) x all 512 columns via
// V_WMMA_F32_16X16X4_F32 (full fp32 matrix path).
//
// A fragment (16x4 f32): float2 at x[row0+(lane&15)][k + 2*(lane>=16)]
// B fragment (4x16 f32): float2 at W[n0+(lane&15)][k + 2*(lane>=16)]
// C/D (16x16 f32): VGPR j = row j (lanes 0-15) / row j+8 (lanes 16-31).
//
// Inner loop: two independent accumulator chains (2 N-tiles) and explicit
// 16-deep register buffers per chain, double-buffered over the two K-halves,
// so B loads run ~32 ahead of their consuming WMMA.
// ---------------------------------------------------------------------------
__global__ __launch_bounds__(256) void at_gemm_kernel(
    const float* __restrict__ x,
    const float* __restrict__ W,
    const float* __restrict__ bias,
    float* __restrict__ h)
{
    const int chunk = blockIdx.x;
    const int wave  = threadIdx.x >> 5;
    const int lane  = threadIdx.x & 31;
    const int half  = lane >> 4;     // 0: lanes 0-15, 1: lanes 16-31
    const int lm    = lane & 15;
    const int row0  = chunk * VBATCH + wave * 16;

    // Preload the wave's 16x128 A panel as 32 K-step fragments (64 VGPRs).
    // x is streamed exactly once -> non-temporal, don't pollute L2.
    const float* xp = x + (size_t)(row0 + lm) * K_IN + 2 * half;
    v2f a[32];
#pragma unroll
    for (int ks = 0; ks < 32; ++ks)
        a[ks] = __builtin_nontemporal_load((const v2f*)(xp + ks * 4));

    for (int nt = 0; nt < 32; nt += 2) {
        const int n0 = nt * 16 + lm;
        const int n1 = n0 + 16;
        const float* wp0 = W + (size_t)n0 * K_IN + 2 * half;   // W stays cached (RT)
        const float* wp1 = W + (size_t)n1 * K_IN + 2 * half;
        const float bias0 = bias[n0];
        const float bias1 = bias[n1];

        v8f c0 = {};
        v8f c1 = {};
        v2f b0[16], b1[16];

        // ---- K half 0: ks = 0..15 ----
#pragma unroll
        for (int ks = 0; ks < 16; ++ks) {
            b0[ks] = *(const v2f*)(wp0 + ks * 4);
            b1[ks] = *(const v2f*)(wp1 + ks * 4);
        }
#pragma unroll
        for (int ks = 0; ks < 16; ++ks) {
            c0 = __builtin_amdgcn_wmma_f32_16x16x4_f32(
                false, a[ks], false, b0[ks], (short)0, c0, false, false);
            c1 = __builtin_amdgcn_wmma_f32_16x16x4_f32(
                false, a[ks], false, b1[ks], (short)0, c1, false, false);
        }

        // ---- K half 1: ks = 16..31 ----
#pragma unroll
        for (int ks = 0; ks < 16; ++ks) {
            b0[ks] = *(const v2f*)(wp0 + 64 + ks * 4);
            b1[ks] = *(const v2f*)(wp1 + 64 + ks * 4);
        }
#pragma unroll
        for (int ks = 0; ks < 16; ++ks) {
            c0 = __builtin_amdgcn_wmma_f32_16x16x4_f32(
                false, a[16 + ks], false, b0[ks], (short)0, c0, false, false);
            c1 = __builtin_amdgcn_wmma_f32_16x16x4_f32(
                false, a[16 + ks], false, b1[ks], (short)0, c1, false, false);
        }

#pragma unroll
        for (int j = 0; j < 8; ++j) {
            const int row = row0 + j + 8 * half;
            h[(size_t)row * N_OUT + n0] = c0[j] + bias0;   // RT: must stay in L2
            h[(size_t)row * N_OUT + n1] = c1[j] + bias1;
        }
    }
}

// ---------------------------------------------------------------------------
// Kernel 2: ghost-BN (training stats, biased var) + sparsemax + prior mult.
// Block = one 128-row virtual batch; reads h (L2-hot) from d_out, overwrites
// d_out in place. Sparsemax tau via Michelot's finite simplex-projection
// iteration (support shrinks monotonically; max element always in support;
// identical tau to the reference's sort-based algorithm, which is
// shift-invariant so the max-subtraction is dropped).
// ---------------------------------------------------------------------------
__device__ __forceinline__ float wave_sum32(float v) {
#pragma unroll
    for (int m = 16; m >= 1; m >>= 1)
        v += __shfl_xor(v, m, 32);
    return v;
}

__global__ __launch_bounds__(256) void at_bn_sparsemax_kernel(
    const float* __restrict__ prior,
    const float* __restrict__ gamma,
    const float* __restrict__ beta,
    float* __restrict__ h)   // in: raw h, out: final result
{
    __shared__ float s_scale[N_OUT];
    __shared__ float s_shift[N_OUT];

    const int chunk  = blockIdx.x;
    const int tid    = threadIdx.x;
    const size_t cb  = (size_t)chunk * VBATCH * N_OUT;

    // ---- Phase A: per-column mean / biased variance over 128 rows ----
#pragma unroll
    for (int cc = 0; cc < 2; ++cc) {
        const int c = tid + cc * 256;
        float s = 0.f, sq = 0.f;
        for (int r = 0; r < VBATCH; ++r) {
            const float v = h[cb + (size_t)r * N_OUT + c];
            s  += v;
            sq += v * v;
        }
        const float mean = s * (1.0f / VBATCH);
        const float var  = sq * (1.0f / VBATCH) - mean * mean;
        const float sc   = gamma[c] * rsqrtf(var + 1e-5f);
        s_scale[c] = sc;
        s_shift[c] = beta[c] - mean * sc;
    }
    __syncthreads();

    // ---- Phase B: per-row sparsemax (one wave per row, 16 rows/wave) ----
    const int wave = tid >> 5;
    const int lane = tid & 31;

    for (int rr = 0; rr < 16; ++rr) {
        const int row = wave * 16 + rr;
        const size_t rbase = cb + (size_t)row * N_OUT;

        float z[16];
        float psum = 0.f;
#pragma unroll
        for (int j = 0; j < 16; ++j) {
            const int c = lane + 32 * j;
            z[j] = h[rbase + c] * s_scale[c] + s_shift[c];
            psum += z[j];
        }

        // tau0 from full support, then Michelot iterations.
        float tau = (wave_sum32(psum) - 1.f) * (1.f / N_OUT);
        for (int it = 0; it < 20; ++it) {
            float ps = 0.f, pc = 0.f;
#pragma unroll
            for (int j = 0; j < 16; ++j) {
                if (z[j] > tau) { ps += z[j]; pc += 1.f; }
            }
            ps = wave_sum32(ps);
            pc = wave_sum32(pc);
            tau = (ps - 1.f) / pc;   // pc >= 1 (max always in support)
        }

        // prior is streamed once (NT load, keep h hot in L2); final output is
        // never re-read (NT store).
#pragma unroll
        for (int j = 0; j < 16; ++j) {
            const int c = lane + 32 * j;
            float m = z[j] - tau;
            m = m > 0.f ? m : 0.f;
            const float p = __builtin_nontemporal_load(prior + rbase + c);
            __builtin_nontemporal_store(m * p, h + rbase + c);
        }
    }
}

// ---------------------------------------------------------------------------
extern "C" void kernel_launch(void* const* d_in, const int* in_sizes, int n_in,
                              void* d_out, int out_size, void* d_ws, size_t ws_size,
                              hipStream_t stream) {
    (void)in_sizes; (void)n_in; (void)d_ws; (void)ws_size; (void)out_size;

    const float* x     = (const float*)d_in[0];   // [65536,128]
    const float* prior = (const float*)d_in[1];   // [65536,512]
    const float* W     = (const float*)d_in[2];   // [512,128]
    const float* bias  = (const float*)d_in[3];   // [512]
    const float* gamma = (const float*)d_in[4];   // [512]
    const float* beta  = (const float*)d_in[5];   // [512]
    float* out = (float*)d_out;                   // [65536,512], also h scratch

    const int nchunks = B_TOTAL / VBATCH;         // 512
    at_gemm_kernel<<<nchunks, 256, 0, stream>>>(x, W, bias, out);
    at_bn_sparsemax_kernel<<<nchunks, 256, 0, stream>>>(prior, gamma, beta, out);
}